// PartialModalityCompensation_46308337386010
// MI455X (gfx1250) — compile-verified
//
#include <hip/hip_runtime.h>
#include <math.h>

typedef float v2f __attribute__((ext_vector_type(2)));
typedef float v8f __attribute__((ext_vector_type(8)));

#define IMGH 512
#define IMGW 512
#define IMGHW (IMGH*IMGW)
#define NB 8
#define FLAGS_D 127

// -------- CDNA5 async global->LDS staging (guarded; falls back to ld+st) --------
#if defined(__HIP_DEVICE_COMPILE__) && __has_builtin(__builtin_amdgcn_global_load_async_to_lds_b32)
#define USE_ASYNC_LDS 1
#else
#define USE_ASYNC_LDS 0
#endif

__device__ __forceinline__ void copy_g2l(const float* src, float* dst) {
#if USE_ASYNC_LDS
    __builtin_amdgcn_global_load_async_to_lds_b32(
        (__attribute__((address_space(1))) int*)(unsigned long long)src,
        (__attribute__((address_space(3))) int*)(unsigned int)(unsigned long long)dst,
        0, 0);
#else
    *dst = *src;
#endif
}

template <int N>
__device__ __forceinline__ void wait_async_le() {
#if USE_ASYNC_LDS
#if __has_builtin(__builtin_amdgcn_s_wait_asynccnt)
    __builtin_amdgcn_s_wait_asynccnt(N);
#else
    asm volatile("s_wait_asynccnt %0" :: "i"(N) : "memory");
#endif
#endif
}

// ---------------- mask pipeline ----------------

__global__ void gray_kernel(const float* __restrict__ rgb, float* __restrict__ gray, int total) {
    int idx = blockIdx.x * blockDim.x + threadIdx.x;
    if (idx >= total) return;
    int b = idx / IMGHW;
    int r = idx - b * IMGHW;
    const float* p = rgb + (size_t)b * 3 * IMGHW + r;
    float g = 0.299f * p[0] + 0.587f * p[IMGHW] + 0.114f * p[2 * IMGHW];
    g = fminf(fmaxf(g * 255.0f, 0.0f), 255.0f);
    gray[idx] = g;
}

__device__ __forceinline__ float gread(const float* g, int y, int x) {
    if (y < 0 || y >= IMGH || x < 0 || x >= IMGW) return 0.0f;
    return g[y * IMGW + x];
}

__global__ void flags_kernel(const float* __restrict__ gray, float* __restrict__ flags, int total) {
    int idx = blockIdx.x * blockDim.x + threadIdx.x;
    if (idx >= total) return;
    int b  = idx / (FLAGS_D * FLAGS_D);
    int r  = idx - b * (FLAGS_D * FLAGS_D);
    int wy = r / FLAGS_D;
    int wx = r - wy * FLAGS_D;
    const float* g = gray + (size_t)b * IMGHW;
    float s1 = 0.0f, s2 = 0.0f;
    for (int dy = 0; dy < 8; ++dy) {
        int y = wy * 4 + dy;
        for (int dx = 0; dx < 8; ++dx) {
            int x = wx * 4 + dx;
            float lap = gread(g, y - 1, x) + gread(g, y + 1, x) +
                        gread(g, y, x - 1) + gread(g, y, x + 1) - 4.0f * gread(g, y, x);
            s1 += lap;
            s2 += lap * lap;
        }
    }
    float var = (s2 - s1 * s1 * (1.0f / 64.0f)) * (1.0f / 63.0f);
    flags[idx] = (var < 50.0f) ? 1.0f : 0.0f;
}

__global__ void masks_kernel(const float* __restrict__ flags, const float* __restrict__ depth,
                             float* __restrict__ ri, float* __restrict__ di, int h, int st, int total) {
    int idx = blockIdx.x * blockDim.x + threadIdx.x;
    if (idx >= total) return;
    int b  = idx / (h * h);
    int r  = idx - b * h * h;
    int yf = r / h;
    int xf = r - yf * h;
    int y = yf * st, x = xf * st;
    const float* f = flags + (size_t)b * FLAGS_D * FLAGS_D;
    float m = 0.0f;
    int wy0 = y >> 2, wx0 = x >> 2;
    for (int a = 0; a < 2; ++a) {
        int wy = wy0 - a;
        if (a == 0) { if (y >= FLAGS_D * 4 + 4) continue; }
        else        { if (y < 4) continue; }
        if (wy < 0 || wy >= FLAGS_D) continue;
        for (int c = 0; c < 2; ++c) {
            int wx = wx0 - c;
            if (c == 0) { if (x >= FLAGS_D * 4 + 4) continue; }
            else        { if (x < 4) continue; }
            if (wx < 0 || wx >= FLAGS_D) continue;
            m = fmaxf(m, f[wy * FLAGS_D + wx]);
        }
    }
    ri[idx] = m;
    di[idx] = (depth[(size_t)b * 3 * IMGHW + 2 * IMGHW + y * IMGW + x] <= 0.0f) ? 1.0f : 0.0f;
}

// ---------------- implicit-GEMM 3x3 conv: double-buffered async-LDS WMMA ----------------
// Block (4 waves): 64 output channels x 64 spatial positions.
// Wave wv: co-subtile wv (16 channels), 4 N-subtiles of 16 with 4 accumulators.
// K = tap-major (tap, ci). Per 32-ci chunk: async-prefetch next chunk into alternate LDS buffer,
// s_wait_asynccnt 32 retires the current buffer's copies, consume with WMMA while prefetch flies.
__global__ __launch_bounds__(128) void conv3x3_wmma(
    const float* __restrict__ x, const float* __restrict__ w, float* __restrict__ z,
    int C, int H, int W) {
    __shared__ float Ws[2][64 * 32];   // [buf][co_local][ci_local]
    __shared__ float Xs[2][32 * 64];   // [buf][ci_local][n_local]

    const int HW   = H * W;
    const int tid  = threadIdx.x;
    const int wv   = tid >> 5;
    const int lane = tid & 31;
    const int ln16 = lane & 15;
    const int kb   = (lane >> 4) * 2;      // K-pair held by this half-wave: 0 or 2
    const int n0   = blockIdx.x * 64;
    const int co0  = blockIdx.y * 64;

    // --- filler role ---
    const int nf_l = tid & 63;             // activation column filled by this thread
    const int ciXb = tid >> 6;             // 0..1
    const int ciW  = tid & 31;             // weight ci filled by this thread
    const int coWb = tid >> 5;             // 0..3
    const int nf = n0 + nf_l;
    const int bF = nf / HW;
    const int rF = nf - bF * HW;
    const int yF = rF / W;
    const int xFc = rF - yF * W;
    const float* xbF = x + (size_t)bF * C * HW;

    // --- consumer role: 4 spatial subtiles ---
    int bC[4], rC[4], yC[4], xC[4];
#pragma unroll
    for (int s = 0; s < 4; ++s) {
        int n = n0 + s * 16 + ln16;
        bC[s] = n / HW;
        rC[s] = n - bC[s] * HW;
        yC[s] = rC[s] / W;
        xC[s] = rC[s] - yC[s] * W;
    }
    const int coL = wv * 16 + ln16;        // block-local A-fragment row

    v8f acc[4] = {};

    // prefetch cursor (next chunk to stage)
    int ftap = 0, fci0 = 0;
    auto do_fill = [&](int buf) {
        const int tap = ftap;
        const int ky = tap / 3 - 1;
        const int kx = tap - (tap / 3) * 3 - 1;
        const int yyF = min(max(yF + ky, 0), H - 1);
        const int xxF = min(max(xFc + kx, 0), W - 1);
        const float* srcBase = xbF + (size_t)yyF * W + xxF;
        const float* wtap = w + tap;
#pragma unroll
        for (int cc = 0; cc < 16; ++cc) {
            int ciX = ciXb + cc * 2;
            copy_g2l(srcBase + (size_t)(fci0 + ciX) * HW, &Xs[buf][ciX * 64 + nf_l]);
        }
#pragma unroll
        for (int cc = 0; cc < 16; ++cc) {
            int coW = coWb + cc * 4;
            copy_g2l(wtap + (size_t)(co0 + coW) * (C * 9) + (size_t)(fci0 + ciW) * 9,
                     &Ws[buf][coW * 32 + ciW]);
        }
        fci0 += 32;
        if (fci0 >= C) { fci0 = 0; ftap++; }
    };

    // prologue: stage chunk 0 into buffer 0
    do_fill(0);
    int buf = 0;

    for (int tap = 0; tap < 9; ++tap) {
        const int ky = tap / 3 - 1;
        const int kx = tap - (tap / 3) * 3 - 1;
        // consumer: validity mask per subtile (zero-pad applied at consumption)
        float vm[4];
#pragma unroll
        for (int s = 0; s < 4; ++s) {
            int yy = yC[s] + ky, xx = xC[s] + kx;
            vm[s] = (yy >= 0 && yy < H && xx >= 0 && xx < W) ? 1.0f : 0.0f;
        }

        for (int ci0 = 0; ci0 < C; ci0 += 32) {
            __syncthreads();                     // all waves done consuming buf^1
            const bool more = (ftap < 9);        // uniform across block
            if (more) {
                do_fill(buf ^ 1);                // 32 async copies for next chunk
                wait_async_le<32>();             // in-order: current buffer's 32 retired
            } else {
                wait_async_le<0>();              // last chunk: drain
            }
            __syncthreads();
            // consume current buffer: 8 K-steps x 4 subtiles
#pragma unroll
            for (int k0 = 0; k0 < 32; k0 += 4) {
                v2f af = *(const v2f*)&Ws[buf][coL * 32 + k0 + kb];
#pragma unroll
                for (int s = 0; s < 4; ++s) {
                    v2f bfv;
                    bfv.x = Xs[buf][(k0 + kb) * 64 + s * 16 + ln16] * vm[s];
                    bfv.y = Xs[buf][(k0 + kb + 1) * 64 + s * 16 + ln16] * vm[s];
                    acc[s] = __builtin_amdgcn_wmma_f32_16x16x4_f32(
                        false, af, false, bfv, (short)0, acc[s], false, false);
                }
            }
            buf ^= 1;
        }
    }
    // store D: VGPR v holds M = v + 8*(lane>=16), N = lane&15
    const int mh = 8 * (lane >> 4);
#pragma unroll
    for (int s = 0; s < 4; ++s) {
        float* zp = z + (size_t)bC[s] * C * HW + rC[s];
#pragma unroll
        for (int v = 0; v < 8; ++v) {
            int com = co0 + wv * 16 + mh + v;
            zp[(size_t)com * HW] = acc[s][v];
        }
    }
}

// ---------------- BN stats: per-channel sum & sumsq over (B,H,W) ----------------
__global__ void bn_stats_kernel(const float* __restrict__ z, float* __restrict__ sums,
                                int C, int HW) {
    int c = blockIdx.x;
    int tid = threadIdx.x;
    float s = 0.0f, s2 = 0.0f;
    for (int b = 0; b < NB; ++b) {
        const float* p = z + ((size_t)b * C + c) * HW;
        for (int i = tid; i < HW; i += blockDim.x) {
            float v = p[i];
            s += v; s2 += v * v;
        }
    }
    __shared__ float sh[256], sh2[256];
    sh[tid] = s; sh2[tid] = s2;
    __syncthreads();
    for (int st = 128; st > 0; st >>= 1) {
        if (tid < st) { sh[tid] += sh[tid + st]; sh2[tid] += sh2[tid + st]; }
        __syncthreads();
    }
    if (tid == 0) { sums[c] = sh[0]; sums[C + c] = sh2[0]; }
}

// ---------------- global average pool of main: gap[b][c] ----------------
__global__ void gap_kernel(const float* __restrict__ m, float* __restrict__ gap, int HW) {
    int bc = blockIdx.x;
    int tid = threadIdx.x;
    const float* p = m + (size_t)bc * HW;
    float s = 0.0f;
    for (int i = tid; i < HW; i += blockDim.x) s += p[i];
    __shared__ float sh[128];
    sh[tid] = s;
    __syncthreads();
    for (int st = 64; st > 0; st >>= 1) {
        if (tid < st) sh[tid] += sh[tid + st];
        __syncthreads();
    }
    if (tid == 0) gap[bc] = sh[0] / (float)HW;
}

// ---------------- channel attention MLP: cw[b][c] ----------------
__global__ void ca_kernel(const float* __restrict__ gap, const float* __restrict__ w1,
                          const float* __restrict__ w2, float* __restrict__ cw, int C, int r) {
    int b = blockIdx.x;
    int tid = threadIdx.x;
    __shared__ float hid[64];
    for (int j = tid; j < r; j += blockDim.x) {
        float s = 0.0f;
        const float* g = gap + (size_t)b * C;
        const float* wr = w1 + (size_t)j * C;
        for (int c = 0; c < C; ++c) s += wr[c] * g[c];
        hid[j] = fmaxf(s, 0.0f);
    }
    __syncthreads();
    for (int c = tid; c < C; c += blockDim.x) {
        float s = 0.0f;
        const float* wr = w2 + (size_t)c * r;
        for (int j = 0; j < r; ++j) s += wr[j] * hid[j];
        cw[(size_t)b * C + c] = 1.0f / (1.0f + expf(-s));
    }
}

// ---------------- spatial reduce of main: channel mean & max per (b,y,x) ----------------
__global__ void spred_kernel(const float* __restrict__ m, float* __restrict__ spmean,
                             float* __restrict__ spmax, int C, int HW, int total) {
    int idx = blockIdx.x * blockDim.x + threadIdx.x;
    if (idx >= total) return;
    int b = idx / HW;
    int r = idx - b * HW;
    const float* p = m + (size_t)b * C * HW + r;
    float s = 0.0f, mx = -3.402823466e38f;
    for (int c = 0; c < C; ++c) {
        float v = p[(size_t)c * HW];
        s += v;
        mx = fmaxf(mx, v);
    }
    spmean[idx] = s / (float)C;
    spmax[idx]  = mx;
}

// ---------------- 7x7 spatial attention conv + sigmoid ----------------
__global__ void sa_kernel(const float* __restrict__ spmean, const float* __restrict__ spmax,
                          const float* __restrict__ saw, float* __restrict__ sw,
                          int H, int W, int total) {
    int idx = blockIdx.x * blockDim.x + threadIdx.x;
    if (idx >= total) return;
    int b = idx / (H * W);
    int r = idx - b * H * W;
    int y = r / W;
    int x = r - y * W;
    const float* pm = spmean + (size_t)b * H * W;
    const float* px = spmax  + (size_t)b * H * W;
    float acc = 0.0f;
    float invW = 2.0f / (float)(W - 1);
    float invH = 2.0f / (float)(H - 1);
    for (int ky = 0; ky < 7; ++ky) {
        int yy = y + ky - 3;
        if (yy < 0 || yy >= H) continue;
        for (int kx = 0; kx < 7; ++kx) {
            int xx = x + kx - 3;
            if (xx < 0 || xx >= W) continue;
            int ko = ky * 7 + kx;
            float xs = -1.0f + invW * (float)xx;
            float ys = -1.0f + invH * (float)yy;
            acc += saw[ko]       * pm[yy * W + xx];
            acc += saw[49 + ko]  * px[yy * W + xx];
            acc += saw[98 + ko]  * xs;
            acc += saw[147 + ko] * ys;
        }
    }
    sw[idx] = 1.0f / (1.0f + expf(-acc));
}

// ---------------- fused BN-ReLU + attention blend ----------------
__global__ void apply_kernel(const float* __restrict__ mainf, const float* __restrict__ z,
                             const float* __restrict__ sums, const float* __restrict__ gma,
                             const float* __restrict__ bta, const float* __restrict__ mask,
                             const float* __restrict__ sw, const float* __restrict__ cw,
                             float* __restrict__ out, int C, int HW, int total, float invN) {
    int idx = blockIdx.x * blockDim.x + threadIdx.x;
    if (idx >= total) return;
    int b   = idx / (C * HW);
    int rem = idx - b * C * HW;
    int c   = rem / HW;
    int r   = rem - c * HW;
    float mean = sums[c] * invN;
    float var  = sums[C + c] * invN - mean * mean;
    float zt   = gma[c] * (z[idx] - mean) * rsqrtf(var + 1e-5f) + bta[c];
    float at   = fmaxf(zt, 0.0f);
    float m    = mask[(size_t)b * HW + r];
    float comp = at * sw[(size_t)b * HW + r] * cw[(size_t)b * C + c] * (1.0f + m);
    out[idx] = mainf[idx] * (1.0f - m) + comp * m;
}

// ---------------- host launcher ----------------
extern "C" void kernel_launch(void* const* d_in, const int* in_sizes, int n_in,
                              void* d_out, int out_size, void* d_ws, size_t ws_size,
                              hipStream_t stream) {
    (void)in_sizes; (void)n_in; (void)out_size; (void)ws_size;
    const float* rgb_feat[3] = {(const float*)d_in[0], (const float*)d_in[2], (const float*)d_in[4]};
    const float* dep_feat[3] = {(const float*)d_in[1], (const float*)d_in[3], (const float*)d_in[5]};
    const float* rgb_img = (const float*)d_in[6];
    const float* dep_img = (const float*)d_in[7];
    const float *ftw[3], *bng[3], *bnb[3], *saw[3], *ca1[3], *ca2[3];
    for (int i = 0; i < 3; ++i) {
        int base = 8 + 6 * i;
        ftw[i] = (const float*)d_in[base + 0];
        bng[i] = (const float*)d_in[base + 1];
        bnb[i] = (const float*)d_in[base + 2];
        saw[i] = (const float*)d_in[base + 3];
        ca1[i] = (const float*)d_in[base + 4];
        ca2[i] = (const float*)d_in[base + 5];
    }

    const int Cs[3] = {256, 512, 1024};
    const int hs[3] = {64, 32, 16};

    float* ws = (float*)d_ws;
    size_t o = 0;
    float* gray = ws + o;  o += (size_t)NB * IMGHW;
    float* flags = ws + o; o += (size_t)NB * FLAGS_D * FLAGS_D;
    float *ri[3], *di[3];
    for (int s = 0; s < 3; ++s) { ri[s] = ws + o; o += (size_t)NB * hs[s] * hs[s]; }
    for (int s = 0; s < 3; ++s) { di[s] = ws + o; o += (size_t)NB * hs[s] * hs[s]; }
    float* zbuf  = ws + o; o += (size_t)NB * 256 * 64 * 64;
    float* bnsum = ws + o; o += 2 * 1024;
    float* gap   = ws + o; o += (size_t)NB * 1024;
    float* cw    = ws + o; o += (size_t)NB * 1024;
    float* spmean = ws + o; o += (size_t)NB * 64 * 64;
    float* spmax  = ws + o; o += (size_t)NB * 64 * 64;
    float* swb    = ws + o; o += (size_t)NB * 64 * 64;

    {
        int tg = NB * IMGHW;
        gray_kernel<<<(tg + 255) / 256, 256, 0, stream>>>(rgb_img, gray, tg);
        int tf = NB * FLAGS_D * FLAGS_D;
        flags_kernel<<<(tf + 255) / 256, 256, 0, stream>>>(gray, flags, tf);
        for (int s = 0; s < 3; ++s) {
            int tm = NB * hs[s] * hs[s];
            masks_kernel<<<(tm + 255) / 256, 256, 0, stream>>>(flags, dep_img, ri[s], di[s],
                                                               hs[s], IMGH / hs[s], tm);
        }
    }

    float* outp = (float*)d_out;
    const size_t off_r[3] = {0ull, 8388608ull, 12582912ull};
    const size_t off_d[3] = {14680064ull, 23068672ull, 27262976ull};

    for (int s = 0; s < 3; ++s) {
        const int C = Cs[s], H = hs[s], W = hs[s], HW = H * W;
        for (int dir = 0; dir < 2; ++dir) {
            const float* mainf = dir ? dep_feat[s] : rgb_feat[s];
            const float* auxf  = dir ? rgb_feat[s] : dep_feat[s];
            const float* mask  = dir ? di[s] : ri[s];
            float* out = outp + (dir ? off_d[s] : off_r[s]);

            dim3 cgrid(NB * HW / 64, C / 64);
            conv3x3_wmma<<<cgrid, 128, 0, stream>>>(auxf, ftw[s], zbuf, C, H, W);
            bn_stats_kernel<<<C, 256, 0, stream>>>(zbuf, bnsum, C, HW);
            gap_kernel<<<NB * C, 128, 0, stream>>>(mainf, gap, HW);
            ca_kernel<<<NB, 256, 0, stream>>>(gap, ca1[s], ca2[s], cw, C, C / 16);
            int tsp = NB * HW;
            spred_kernel<<<(tsp + 255) / 256, 256, 0, stream>>>(mainf, spmean, spmax, C, HW, tsp);
            sa_kernel<<<(tsp + 255) / 256, 256, 0, stream>>>(spmean, spmax, saw[s], swb, H, W, tsp);
            int tot = NB * C * HW;
            apply_kernel<<<(tot + 255) / 256, 256, 0, stream>>>(mainf, zbuf, bnsum, bng[s], bnb[s],
                                                                mask, swb, cw, out, C, HW, tot,
                                                                1.0f / (float)(NB * HW));
        }
    }
}